// MVG_binaryNet_53532472377583
// MI455X (gfx1250) — compile-verified
//
#include <hip/hip_runtime.h>
#include <math.h>

#define SQ2PI 0.79788456f

typedef __attribute__((ext_vector_type(2))) float v2f;
typedef __attribute__((ext_vector_type(8))) float v8f;
typedef __attribute__((ext_vector_type(4))) int   v4i;
typedef __attribute__((address_space(1))) v4i v4i_global_t;   // global (AS1)
typedef __attribute__((address_space(3))) v4i v4i_lds_t;      // LDS (AS3)

__device__ inline v8f wmma_f32(v2f a, v2f b, v8f c) {
  // D = A(16x4) x B(4x16) + C, exact f32 on the CDNA5 matrix pipe.
  return __builtin_amdgcn_wmma_f32_16x16x4_f32(false, a, false, b, (short)0, c, false, false);
}

// ---- CDNA5 async global->LDS copy (ASYNCcnt) with safe fallback -----------
#if defined(__gfx1250__) && __has_builtin(__builtin_amdgcn_global_load_async_to_lds_b128)
#define HAVE_ASYNC_LDS 1
#endif

__device__ inline void async_copy_b128(float* lds_dst, const float* gsrc) {
#if defined(HAVE_ASYNC_LDS)
  v4i* g0 = (v4i*)gsrc;      // drop const + retype in generic AS
  v4i* l0 = (v4i*)lds_dst;
  __builtin_amdgcn_global_load_async_to_lds_b128(
      (v4i_global_t*)g0, (v4i_lds_t*)l0, 0, 0);
#else
  *reinterpret_cast<float4*>(lds_dst) = *reinterpret_cast<const float4*>(gsrc);
#endif
}

__device__ inline void wait_async_all() {
#if defined(HAVE_ASYNC_LDS)
#if __has_builtin(__builtin_amdgcn_s_wait_asynccnt)
  __builtin_amdgcn_s_wait_asynccnt(0);
#else
  asm volatile("s_wait_asynccnt 0x0" ::: "memory");
#endif
#endif
}

// ---------------------------------------------------------------------------
// Batched 512x512x512 GEMM on the WMMA f32 pipe.
//   C_b[m,n] = sum_k A[k,m] * B_b[k,n]     (A consumed as [K,M] rows)
//   All operands are either naturally [K,M]/[K,N] row-major (weight masks,
//   T) or symmetric (xcov2), so every LDS fill is a contiguous B128 row copy
//   -> GLOBAL_LOAD_ASYNC_TO_LDS_B128 + double-buffered chunks.
//   SCALE_A : multiply A rows by scale[b,k] (folds diag(d1) into the GEMM;
//             uses the synchronous fill path since async bypasses VGPRs)
//   BATCH_A / BATCH_B : operand has a 512*512 per-batch stride
//   XCOV_EPI: out = SQ2PI*(acc + (m==n)*colsum[m])*xc2[b,m]*xc2[b,n]
//             + (m==n)*(1 - xbar[b,m]^2)
// Block: 256 thr = 8 waves; 128x128 tile; wave = 64x32 (4x2 grid of 16x16).
// ---------------------------------------------------------------------------
template<bool SCALE_A, bool BATCH_A, bool BATCH_B, bool XCOV_EPI>
__global__ __launch_bounds__(256) void gemm512_wmma_kernel(
    const float* __restrict__ A, const float* __restrict__ B, float* __restrict__ C,
    const float* __restrict__ scale, const float* __restrict__ colsum,
    const float* __restrict__ xc2, const float* __restrict__ xbar)
{
  constexpr int NN = 512, BK = 16, NCH = NN / BK;
  __shared__ float aS[2][BK][132];   // aS[buf][k][m]
  __shared__ float bS[2][BK][132];   // bS[buf][k][n]

  const int b  = blockIdx.z;
  const int m0 = blockIdx.y * 128;
  const int n0 = blockIdx.x * 128;
  const size_t matOff = (size_t)b * NN * NN;
  const float* Ab = A + (BATCH_A ? matOff : 0);
  const float* Bb = B + (BATCH_B ? matOff : 0);

  const int t    = threadIdx.x;
  const int lane = t & 31;
  const int wave = t >> 5;
  const int wm0  = (wave & 1) * 64;   // wave row offset inside block tile
  const int wn0  = (wave >> 1) * 32;  // wave col offset inside block tile
  const int lm   = lane & 15;
  const int kb   = (lane >> 4) << 1;  // lanes 0-15 -> K{0,1}, 16-31 -> K{2,3}
  const int rhi  = (lane >> 4) << 3;  // C/D: hi lanes hold M+8

  auto fill = [&](int ci, int bi) {
    const int k0 = ci * BK;
    #pragma unroll
    for (int i = 0; i < 2; i++) {
      int e = t + i * 256;            // 0..511 float4s (16 rows x 32 f4)
      int k = e >> 5;
      int c = (e & 31) << 2;
      const float* gp = Ab + (size_t)(k0 + k) * NN + m0 + c;
      if (SCALE_A) {
        float s = scale[(size_t)b * NN + k0 + k];
        float4 v = *reinterpret_cast<const float4*>(gp);
        aS[bi][k][c + 0] = v.x * s; aS[bi][k][c + 1] = v.y * s;
        aS[bi][k][c + 2] = v.z * s; aS[bi][k][c + 3] = v.w * s;
      } else {
        async_copy_b128(&aS[bi][k][c], gp);
      }
    }
    #pragma unroll
    for (int i = 0; i < 2; i++) {
      int e = t + i * 256;
      int k = e >> 5;
      int c = (e & 31) << 2;
      async_copy_b128(&bS[bi][k][c], Bb + (size_t)(k0 + k) * NN + n0 + c);
    }
  };

  v8f acc[4][2];
  #pragma unroll
  for (int i = 0; i < 4; i++)
    #pragma unroll
    for (int j = 0; j < 2; j++)
      #pragma unroll
      for (int r = 0; r < 8; r++) acc[i][j][r] = 0.f;

  fill(0, 0);
  wait_async_all();
  __syncthreads();

  for (int ci = 0; ci < NCH; ++ci) {
    const int cur = ci & 1;
    if (ci + 1 < NCH) fill(ci + 1, cur ^ 1);   // prefetch next chunk (async)

    #pragma unroll
    for (int kk = 0; kk < BK; kk += 4) {
      v2f af[4], bf[2];
      #pragma unroll
      for (int im = 0; im < 4; im++) {
        int mm = wm0 + im * 16 + lm;
        af[im].x = aS[cur][kk + kb + 0][mm];
        af[im].y = aS[cur][kk + kb + 1][mm];
      }
      #pragma unroll
      for (int in_ = 0; in_ < 2; in_++) {
        int nn = wn0 + in_ * 16 + lm;
        bf[in_].x = bS[cur][kk + kb + 0][nn];
        bf[in_].y = bS[cur][kk + kb + 1][nn];
      }
      #pragma unroll
      for (int im = 0; im < 4; im++)
        #pragma unroll
        for (int in_ = 0; in_ < 2; in_++)
          acc[im][in_] = wmma_f32(af[im], bf[in_], acc[im][in_]);
    }

    wait_async_all();    // next chunk landed in LDS
    __syncthreads();     // everyone done reading cur + next buffer visible
  }

  float* Cb = C + matOff;
  #pragma unroll
  for (int im = 0; im < 4; im++) {
    #pragma unroll
    for (int in_ = 0; in_ < 2; in_++) {
      int gn = n0 + wn0 + in_ * 16 + lm;
      #pragma unroll
      for (int r = 0; r < 8; r++) {
        int gm = m0 + wm0 + im * 16 + rhi + r;
        float val = acc[im][in_][r];
        if (XCOV_EPI) {
          if (gm == gn) val += colsum[gm];
          val = SQ2PI * val * xc2[(size_t)b * 512 + gm] * xc2[(size_t)b * 512 + gn];
          if (gm == gn) {
            float xb = xbar[(size_t)b * 512 + gm];
            val += 1.f - xb * xb;
          }
        }
        Cb[(size_t)gm * 512 + gn] = val;
      }
    }
  }
}

// ---------------------------------------------------------------------------
// Small elementwise / reduction kernels (all tiny vs the GEMMs).
// ---------------------------------------------------------------------------
__global__ void mask_kernel(const float* __restrict__ w, float* __restrict__ m, int n) {
  int i = blockIdx.x * blockDim.x + threadIdx.x;
  if (i < n) {
    float s = 1.f / (1.f + expf(-w[i]));
    m[i] = 2.f * s - 1.f;
  }
}

__global__ void colsum_kernel(const float* __restrict__ m, float* __restrict__ cs, int rows) {
  int h = blockIdx.x * blockDim.x + threadIdx.x;
  if (h < 512) {
    float s = 0.f;
    for (int r = 0; r < rows; r++) { float v = m[(size_t)r * 512 + h]; s += 1.f - v * v; }
    cs[h] = s;
  }
}

__global__ void layer1_kernel(const float* __restrict__ x, const float* __restrict__ m0,
                              const float* __restrict__ th0, const float* __restrict__ cs0,
                              float* __restrict__ x1bar, float* __restrict__ d1) {
  int i = blockIdx.x * blockDim.x + threadIdx.x;   // 128*512
  int b = i >> 9, h = i & 511;
  float s = th0[h];
  const float* xr = x + (size_t)b * 784;
  for (int j = 0; j < 784; j++) s += xr[j] * m0[(size_t)j * 512 + h];
  float r = rsqrtf(cs0[h]);
  float tt = tanhf(SQ2PI * s * r);
  x1bar[i] = tt;
  d1[i] = 1.f - tt * tt;
}

// dsig[b,h] = cs[h] + sum_j d1[b,j]*m[j,h]^2   (layer 2: xcov_1 is diagonal)
__global__ void diagsig_from_diag_kernel(const float* __restrict__ d1, const float* __restrict__ m,
                                         const float* __restrict__ cs, float* __restrict__ dsig) {
  int i = blockIdx.x * blockDim.x + threadIdx.x;
  int b = i >> 9, h = i & 511;
  float s = cs[h];
  const float* dr = d1 + (size_t)b * 512;
  for (int j = 0; j < 512; j++) { float mm = m[(size_t)j * 512 + h]; s += dr[j] * mm * mm; }
  dsig[i] = s;
}

// dsig[b,h] = cs[h] + sum_j m[j,h] * T[b,j,h]   (diag of m^T xcov m via T=xcov*m)
__global__ void diagsig_from_T_kernel(const float* __restrict__ m, const float* __restrict__ T,
                                      const float* __restrict__ cs, float* __restrict__ dsig) {
  int i = blockIdx.x * blockDim.x + threadIdx.x;
  int b = i >> 9, h = i & 511;
  float s = cs[h];
  const float* Tb = T + (size_t)b * 512 * 512;
  for (int j = 0; j < 512; j++) s += m[(size_t)j * 512 + h] * Tb[(size_t)j * 512 + h];
  dsig[i] = s;
}

__global__ void act_kernel(const float* __restrict__ xprev, const float* __restrict__ m,
                           const float* __restrict__ th, const float* __restrict__ dsig,
                           float* __restrict__ xbar, float* __restrict__ xc2) {
  int i = blockIdx.x * blockDim.x + threadIdx.x;
  int b = i >> 9, h = i & 511;
  float s = th[h];
  const float* xr = xprev + (size_t)b * 512;
  for (int j = 0; j < 512; j++) s += xr[j] * m[(size_t)j * 512 + h];
  float r = rsqrtf(dsig[i]);
  float tt = tanhf(SQ2PI * s * r);
  xbar[i] = tt;
  xc2[i] = (1.f - tt * tt) * r;
}

__global__ void head_kernel(const float* __restrict__ x4bar, const float* __restrict__ mlast,
                            const float* __restrict__ thlast, const int* __restrict__ target,
                            float* __restrict__ out_hlast, float* __restrict__ out_logp,
                            float* __restrict__ out_loss, float* __restrict__ out_frac) {
  __shared__ float sl[128];
  __shared__ float sc[128];
  int b = threadIdx.x;     // 128 threads, one per batch row
  float h[10];
  const float* xr = x4bar + (size_t)b * 512;
  for (int o = 0; o < 10; o++) {
    float s = thlast[o];
    for (int j = 0; j < 512; j++) s += xr[j] * mlast[(size_t)j * 10 + o];
    h[o] = s;
    out_hlast[b * 10 + o] = s;
  }
  float mx = h[0]; int am = 0;
  for (int o = 1; o < 10; o++) if (h[o] > mx) { mx = h[o]; am = o; }  // first-max ties
  float se = 0.f;
  for (int o = 0; o < 10; o++) se += expf(h[o] - mx);
  float lse = mx + logf(se);
  for (int o = 0; o < 10; o++) out_logp[b * 10 + o] = h[o] - lse;
  int tg = target[b];
  sl[b] = -(h[tg] - lse);
  sc[b] = (am == tg) ? 1.f : 0.f;
  __syncthreads();
  for (int s2 = 64; s2 > 0; s2 >>= 1) {
    if (b < s2) { sl[b] += sl[b + s2]; sc[b] += sc[b + s2]; }
    __syncthreads();
  }
  if (b == 0) { *out_loss = sl[0] / 128.f; *out_frac = sc[0] / 128.f; }
}

// ---------------------------------------------------------------------------
extern "C" void kernel_launch(void* const* d_in, const int* in_sizes, int n_in,
                              void* d_out, int out_size, void* d_ws, size_t ws_size,
                              hipStream_t stream) {
  (void)in_sizes; (void)n_in; (void)out_size; (void)ws_size;
  const float* x      = (const float*)d_in[0];
  const int*   target = (const int*)d_in[1];
  const float* w0     = (const float*)d_in[2];
  const float* w1     = (const float*)d_in[3];
  const float* w2     = (const float*)d_in[4];
  const float* w3     = (const float*)d_in[5];
  const float* wlast  = (const float*)d_in[6];
  const float* th0    = (const float*)d_in[7];
  const float* th1    = (const float*)d_in[8];
  const float* th2    = (const float*)d_in[9];
  const float* th3    = (const float*)d_in[10];
  const float* thlast = (const float*)d_in[11];

  float* ws = (float*)d_ws;
  size_t off = 0;
  auto alloc = [&](size_t n) { float* p = ws + off; off += n; return p; };
  float* m0    = alloc((size_t)784 * 512);
  float* m1    = alloc((size_t)512 * 512);
  float* m2    = alloc((size_t)512 * 512);
  float* m3    = alloc((size_t)512 * 512);
  float* mlast = alloc((size_t)512 * 10);
  float* cs0 = alloc(512); float* cs1 = alloc(512);
  float* cs2 = alloc(512); float* cs3 = alloc(512);
  float* x1bar = alloc(128 * 512); float* x2bar = alloc(128 * 512);
  float* x3bar = alloc(128 * 512); float* x4bar = alloc(128 * 512);
  float* d1    = alloc(128 * 512);
  float* xc2_2 = alloc(128 * 512); float* xc2_3 = alloc(128 * 512);
  float* xc2_4 = alloc(128 * 512);
  float* dsig2 = alloc(128 * 512); float* dsig3 = alloc(128 * 512);
  float* dsig4 = alloc(128 * 512);
  float* xcov2 = alloc((size_t)128 * 512 * 512);
  float* T     = alloc((size_t)128 * 512 * 512);

  float* out        = (float*)d_out;
  float* out_hlast  = out;                               // [128,10]
  float* out_logp   = out + 1280;                        // [128,10]
  float* out_xcov3  = out + 2560;                        // [128,512,512]
  float* out_loss   = out + 2560 + (size_t)128 * 512 * 512;
  float* out_frac   = out_loss + 1;

  const dim3 g4(4, 4, 128);

  // Masks m = 2*sigmoid(w)-1 and column sums of (1-m^2).
  mask_kernel<<<(784 * 512 + 255) / 256, 256, 0, stream>>>(w0, m0, 784 * 512);
  mask_kernel<<<(512 * 512 + 255) / 256, 256, 0, stream>>>(w1, m1, 512 * 512);
  mask_kernel<<<(512 * 512 + 255) / 256, 256, 0, stream>>>(w2, m2, 512 * 512);
  mask_kernel<<<(512 * 512 + 255) / 256, 256, 0, stream>>>(w3, m3, 512 * 512);
  mask_kernel<<<(512 * 10 + 255) / 256, 256, 0, stream>>>(wlast, mlast, 512 * 10);
  colsum_kernel<<<2, 256, 0, stream>>>(m0, cs0, 784);
  colsum_kernel<<<2, 256, 0, stream>>>(m1, cs1, 512);
  colsum_kernel<<<2, 256, 0, stream>>>(m2, cs2, 512);
  colsum_kernel<<<2, 256, 0, stream>>>(m3, cs3, 512);

  // Layer 1 (diagonal xcov_1 = diag(d1)).
  layer1_kernel<<<256, 256, 0, stream>>>(x, m0, th0, cs0, x1bar, d1);

  // Layer 2: diagsig analytically, activation, then one fused GEMM -> xcov_2.
  // sigma2 = (d1 . m1)^T m1 : A = m1 [K,M] with per-k scaling.
  diagsig_from_diag_kernel<<<256, 256, 0, stream>>>(d1, m1, cs1, dsig2);
  act_kernel<<<256, 256, 0, stream>>>(x1bar, m1, th1, dsig2, x2bar, xc2_2);
  gemm512_wmma_kernel<true, false, false, true><<<g4, 256, 0, stream>>>(
      m1, m1, xcov2, d1, cs1, xc2_2, x2bar);

  // Layer 3: T = xcov2 @ m2 (xcov2 symmetric -> read rows as [K,M]);
  // diag; act; sigma3 = m2^T @ T -> xcov_3 (returned output).
  gemm512_wmma_kernel<false, true, false, false><<<g4, 256, 0, stream>>>(
      xcov2, m2, T, nullptr, nullptr, nullptr, nullptr);
  diagsig_from_T_kernel<<<256, 256, 0, stream>>>(m2, T, cs2, dsig3);
  act_kernel<<<256, 256, 0, stream>>>(x2bar, m2, th2, dsig3, x3bar, xc2_3);
  gemm512_wmma_kernel<false, false, true, true><<<g4, 256, 0, stream>>>(
      m2, T, out_xcov3, nullptr, cs2, xc2_3, x3bar);

  // Layer 4 (uses xcov_2, faithful to reference; xcov_4 never materialized):
  // only diag(m3^T xcov2 m3) is required -> one GEMM U = xcov2 @ m3.
  gemm512_wmma_kernel<false, true, false, false><<<g4, 256, 0, stream>>>(
      xcov2, m3, T, nullptr, nullptr, nullptr, nullptr);
  diagsig_from_T_kernel<<<256, 256, 0, stream>>>(m3, T, cs3, dsig4);
  act_kernel<<<256, 256, 0, stream>>>(x3bar, m3, th3, dsig4, x4bar, xc2_4);

  // Head: hlastbar, log_softmax, loss, fraction correct.
  head_kernel<<<1, 128, 0, stream>>>(x4bar, mlast, thlast, target,
                                     out_hlast, out_logp, out_loss, out_frac);
}